// Time_derivative_diffusion_product_83854941487389
// MI455X (gfx1250) — compile-verified
//
#include <hip/hip_runtime.h>
#include <hip/hip_bf16.h>
#include <math.h>

// Shapes from the reference
#define V0N 64
#define V1N 128
#define VN  8192   // V0N * V1N
#define CN  128

#define KC   32                 // K-chunk staged in LDS per block
#define LDSF (KC * CN)          // floats per LDS buffer (4096 = 16 KB)

typedef __attribute__((ext_vector_type(2))) float v2f;
typedef __attribute__((ext_vector_type(8))) float v8f;

// LDS layout for the B panel (interleaved row pairs):
//   float index of B[k][c] within a chunk = ((k>>1)*CN + c)*2 + (k&1)
// so {B[k][c], B[k+1][c]} for even k is one aligned 8-byte pair -> ds_load_b64,
// and staging writes {B[2p][c],B[2p+1][c],B[2p][c+1],B[2p+1][c+1]} as one
// ds_store_b128.

// ---------------------------------------------------------------------------
// Kernel 1: y[k,c] = coef(k,c) * sum_i E[i,k] * x[i,c] * mass[i]
// Block = 4 waves, each wave owns a 16-row (k) x 128-col (c) output strip.
// B panel (x ⊙ mass) is staged cooperatively into LDS, double buffered.
// ---------------------------------------------------------------------------
__global__ __launch_bounds__(128) void kspec_project(
    const float* __restrict__ E,      // evecs [VN x VN], row major
    const float* __restrict__ x,      // [VN x CN]
    const float* __restrict__ mass,   // [VN]
    const float* __restrict__ ev0,    // [V0N]
    const float* __restrict__ ev1,    // [V1N]
    const float* __restrict__ dt,     // [2 x CN]
    float* __restrict__ y)            // [VN x CN] workspace output
{
  __shared__ float sbuf[2 * LDSF];    // 32 KB double buffer

  const int lane = threadIdx.x;        // 0..31
  const int lh   = lane >> 4;          // half-wave
  const int lm   = lane & 15;
  const int tid  = threadIdx.y * 32 + lane;              // 0..127
  const int strip = blockIdx.x * blockDim.y + threadIdx.y;
  const int m0 = strip * 16;           // spectral-row base (k of output)

  v8f acc[8] = {};

  // --- stage chunk 0 ---
#pragma unroll
  for (int j = 0; j < 8; ++j) {
    const int q  = tid + j * 128;      // 0..1023
    const int p  = q >> 6;             // row pair within chunk
    const int cp = q & 63;             // column pair
    const int c  = cp * 2;
    const int k0 = 2 * p;              // chunk base 0
    const v2f g0 = *(const v2f*)(x + (size_t)(k0 + 0) * CN + c);
    const v2f g1 = *(const v2f*)(x + (size_t)(k0 + 1) * CN + c);
    const float ms0 = mass[k0 + 0], ms1 = mass[k0 + 1];
    float4 w;
    w.x = g0.x * ms0; w.y = g1.x * ms1; w.z = g0.y * ms0; w.w = g1.y * ms1;
    *(float4*)(sbuf + p * 256 + cp * 4) = w;
  }
  __syncthreads();

  for (int kc = 0; kc < VN; kc += KC) {
    const int cur = (kc >> 5) & 1;
    float* buf  = sbuf + cur * LDSF;
    float* nbuf = sbuf + (1 - cur) * LDSF;

    // stage next chunk while computing this one
    if (kc + KC < VN) {
#pragma unroll
      for (int j = 0; j < 8; ++j) {
        const int q  = tid + j * 128;
        const int p  = q >> 6;
        const int cp = q & 63;
        const int c  = cp * 2;
        const int k0 = kc + KC + 2 * p;
        const v2f g0 = *(const v2f*)(x + (size_t)(k0 + 0) * CN + c);
        const v2f g1 = *(const v2f*)(x + (size_t)(k0 + 1) * CN + c);
        const float ms0 = mass[k0 + 0], ms1 = mass[k0 + 1];
        float4 w;
        w.x = g0.x * ms0; w.y = g1.x * ms1; w.z = g0.y * ms0; w.w = g1.y * ms1;
        *(float4*)(nbuf + p * 256 + cp * 4) = w;
      }
    }

    // compute 8 WMMA K-steps from the current buffer
#pragma unroll
    for (int s = 0; s < 8; ++s) {
      const int ia = kc + 4 * s + 2 * lh;   // this half-wave's reduction rows
      v2f a;                                // A^T fragment from evecs columns
      a.x = E[(size_t)(ia + 0) * VN + m0 + lm];
      a.y = E[(size_t)(ia + 1) * VN + m0 + lm];
      const float* bp = buf + (2 * s + lh) * 256;   // row-pair base in LDS
#pragma unroll
      for (int t = 0; t < 8; ++t) {
        const v2f b = *(const v2f*)(bp + (t * 16 + lm) * 2);  // ds_load_b64
        acc[t] = __builtin_amdgcn_wmma_f32_16x16x4_f32(
            false, a, false, b, (short)0, acc[t], false, false);
      }
    }
    __syncthreads();
  }

  // Epilogue: per-channel Kronecker diffusion coefficient, then store y.
#pragma unroll
  for (int t = 0; t < 8; ++t) {
    const int c = t * 16 + lm;
    const float t0 = dt[c];
    const float t1 = dt[CN + c];
#pragma unroll
    for (int r = 0; r < 8; ++r) {
      const int k = m0 + r + 8 * lh;
      const float coef = __expf(-(ev0[k >> 7] * t0 + ev1[k & 127] * t1));
      y[(size_t)k * CN + c] = acc[t][r] * coef;
    }
  }
}

// ---------------------------------------------------------------------------
// Kernel 2: out[m,c] = sum_k E[m,k] * y[k,c]
// Same structure; A fragments are per-lane contiguous b64 row streams.
// ---------------------------------------------------------------------------
__global__ __launch_bounds__(128) void kspec_recon(
    const float* __restrict__ E,      // evecs [VN x VN]
    const float* __restrict__ y,      // [VN x CN]
    float* __restrict__ out)          // [VN x CN]
{
  __shared__ float sbuf[2 * LDSF];

  const int lane = threadIdx.x;
  const int lh   = lane >> 4;
  const int lm   = lane & 15;
  const int tid  = threadIdx.y * 32 + lane;
  const int strip = blockIdx.x * blockDim.y + threadIdx.y;
  const int m0 = strip * 16;

  v8f acc[8] = {};

#pragma unroll
  for (int j = 0; j < 8; ++j) {
    const int q  = tid + j * 128;
    const int p  = q >> 6;
    const int cp = q & 63;
    const int c  = cp * 2;
    const int k0 = 2 * p;
    const v2f g0 = *(const v2f*)(y + (size_t)(k0 + 0) * CN + c);
    const v2f g1 = *(const v2f*)(y + (size_t)(k0 + 1) * CN + c);
    float4 w;
    w.x = g0.x; w.y = g1.x; w.z = g0.y; w.w = g1.y;
    *(float4*)(sbuf + p * 256 + cp * 4) = w;
  }
  __syncthreads();

  const float* arow = E + (size_t)(m0 + lm) * VN;   // this lane's A row

  for (int kc = 0; kc < VN; kc += KC) {
    const int cur = (kc >> 5) & 1;
    float* buf  = sbuf + cur * LDSF;
    float* nbuf = sbuf + (1 - cur) * LDSF;

    if (kc + KC < VN) {
      // prefetch the A row stream two chunks ahead (global_prefetch_b8)
      __builtin_prefetch(arow + kc + 2 * KC, 0, 3);
#pragma unroll
      for (int j = 0; j < 8; ++j) {
        const int q  = tid + j * 128;
        const int p  = q >> 6;
        const int cp = q & 63;
        const int c  = cp * 2;
        const int k0 = kc + KC + 2 * p;
        const v2f g0 = *(const v2f*)(y + (size_t)(k0 + 0) * CN + c);
        const v2f g1 = *(const v2f*)(y + (size_t)(k0 + 1) * CN + c);
        float4 w;
        w.x = g0.x; w.y = g1.x; w.z = g0.y; w.w = g1.y;
        *(float4*)(nbuf + p * 256 + cp * 4) = w;
      }
    }

#pragma unroll
    for (int s = 0; s < 8; ++s) {
      const int ka = kc + 4 * s + 2 * lh;
      const v2f a = *(const v2f*)(arow + ka);        // b64 row stream
      const float* bp = buf + (2 * s + lh) * 256;
#pragma unroll
      for (int t = 0; t < 8; ++t) {
        const v2f b = *(const v2f*)(bp + (t * 16 + lm) * 2);
        acc[t] = __builtin_amdgcn_wmma_f32_16x16x4_f32(
            false, a, false, b, (short)0, acc[t], false, false);
      }
    }
    __syncthreads();
  }

#pragma unroll
  for (int t = 0; t < 8; ++t) {
#pragma unroll
    for (int r = 0; r < 8; ++r) {
      out[(size_t)(m0 + r + 8 * lh) * CN + t * 16 + lm] = acc[t][r];
    }
  }
}

extern "C" void kernel_launch(void* const* d_in, const int* in_sizes, int n_in,
                              void* d_out, int out_size, void* d_ws, size_t ws_size,
                              hipStream_t stream) {
  // setup_inputs() order:
  //   0: x [V,C] f32, 1: edge_index (unused), 2: L (unused), 3: mass [V] f32,
  //   4: evals0 [64] f32, 5: evals1 [128] f32, 6: evecs [V,V] f32,
  //   7: diffusion_time [2,C] f32
  const float* x    = (const float*)d_in[0];
  const float* mass = (const float*)d_in[3];
  const float* ev0  = (const float*)d_in[4];
  const float* ev1  = (const float*)d_in[5];
  const float* E    = (const float*)d_in[6];
  const float* dt   = (const float*)d_in[7];
  float* out = (float*)d_out;
  float* y   = (float*)d_ws;                 // 8192*128*4 = 4 MB scratch

  dim3 blk(32, 4);                           // 4 waves / block, wave32
  dim3 grd((VN / 16) / 4);                   // 512 strips / 4 waves = 128 blocks

  kspec_project<<<grd, blk, 0, stream>>>(E, x, mass, ev0, ev1, dt, y);
  kspec_recon  <<<grd, blk, 0, stream>>>(E, y, out);
}